// MultiHeadGlobalMemory_13649406067119
// MI455X (gfx1250) — compile-verified
//
#include <hip/hip_runtime.h>
#include <math.h>

#define NROW 200000
#define DIM 512
#define NH 8
#define HD 64
#define MTILE 64

typedef float v2f __attribute__((ext_vector_type(2)));
typedef float v8f __attribute__((ext_vector_type(8)));
typedef unsigned v4u __attribute__((ext_vector_type(4)));
typedef int v4i __attribute__((ext_vector_type(4)));
typedef int v8i __attribute__((ext_vector_type(8)));

// monotonic float<->uint mapping so atomicMax(uint) == float max
__device__ __forceinline__ unsigned f2ord(float f) {
  unsigned u = __float_as_uint(f);
  return (u & 0x80000000u) ? ~u : (u | 0x80000000u);
}
__device__ __forceinline__ float ord2f(unsigned u) {
  return __uint_as_float((u & 0x80000000u) ? (u ^ 0x80000000u) : ~u);
}

// ---------------- phase 0: init accumulators ----------------
__global__ void k_init(unsigned* gmax, float* gsum, float* G) {
  int t = threadIdx.x;
  if (t < NH) { gmax[t] = 0u; gsum[t] = 0.f; }
  if (t < NH * HD) G[t] = 0.f;
}

// ---------------- phase 1: e[n,h] + per-head max ----------------
__global__ __launch_bounds__(256) void k_e(const float* __restrict__ E,
                                           const float* __restrict__ gmem,
                                           const float* __restrict__ attn,
                                           float* __restrict__ ws_e,
                                           unsigned* __restrict__ gmax) {
  __shared__ float sAh[NH * HD];
  __shared__ float sAg[NH];
  __shared__ unsigned smax[NH];
  int t = threadIdx.x;
  for (int i = t; i < NH * HD; i += 256) {
    int h = i >> 6, d = i & 63;
    sAh[i] = attn[h * 2 * HD + HD + d];
  }
  if (t < NH) {
    float s = 0.f;
    for (int d = 0; d < HD; ++d) s += gmem[t * HD + d] * attn[t * 2 * HD + d];
    sAg[t] = s;
    smax[t] = 0u;
  }
  __syncthreads();
  long long p = (long long)blockIdx.x * 256 + t;
  int n = (int)(p >> 3), h = (int)(p & 7);
  const float* row = E + (long long)n * DIM + h * HD;
  float dot = 0.f;
#pragma unroll 8
  for (int d = 0; d < HD; ++d) dot += row[d] * sAh[h * HD + d];
  float x = dot + sAg[h];
  float e = x >= 0.f ? x : 0.2f * x;
  ws_e[p] = e;
  atomicMax(&smax[h], f2ord(e));
  __syncthreads();
  if (t < NH) atomicMax(&gmax[t], smax[t]);
}

// ---------------- phase 2: sum(exp) and unnormalized g_new ----------------
__global__ __launch_bounds__(256) void k_sum(const float* __restrict__ E,
                                             const float* __restrict__ ws_e,
                                             const unsigned* __restrict__ gmax,
                                             float* __restrict__ gsum,
                                             float* __restrict__ G) {
  __shared__ float sMax[NH];
  __shared__ float ssum[NH];
  __shared__ float sG[NH * HD];
  int t = threadIdx.x;
  int lane = t & 31;
  if (t < NH) { sMax[t] = ord2f(gmax[t]); ssum[t] = 0.f; }
  for (int i = t; i < NH * HD; i += 256) sG[i] = 0.f;
  __syncthreads();
  long long p = (long long)blockIdx.x * 256 + t;
  int n = (int)(p >> 3), h = (int)(p & 7);
  float pv = __expf(ws_e[p] - sMax[h]);
  {  // lanes {h, h+8, h+16, h+24} share head h: fold 4 rows before atomics
    float v = pv;
    v += __shfl_xor(v, 8);
    v += __shfl_xor(v, 16);
    if (lane < 8) atomicAdd(&ssum[h], v);
  }
  const float* row = E + (long long)n * DIM + h * HD;
  for (int d = 0; d < HD; ++d) {
    float v = pv * row[d];
    v += __shfl_xor(v, 8);
    v += __shfl_xor(v, 16);
    if (lane < 8) atomicAdd(&sG[h * HD + d], v);
  }
  __syncthreads();
  if (t < NH) atomicAdd(&gsum[t], ssum[t]);
  for (int i = t; i < NH * HD; i += 256) atomicAdd(&G[i], sG[i]);
}

// ---------------- phase 3: finalize g_new, fold constants into c1 ----------------
__global__ __launch_bounds__(512) void k_final(const float* __restrict__ gsum,
                                               const float* __restrict__ G,
                                               const float* __restrict__ W1,
                                               const float* __restrict__ b1,
                                               float* __restrict__ g_new,
                                               float* __restrict__ c1) {
  __shared__ float sGn[NH * HD];
  int t = threadIdx.x;  // 0..511
  float gn = G[t] / gsum[t >> 6];
  sGn[t] = gn;
  g_new[t] = gn;
  __syncthreads();
  int h = t >> 6, e = t & 63;
  float c = b1[t];
  for (int d = 0; d < HD; ++d)
    c += sGn[h * HD + d] * W1[h * 2 * HD * HD + (HD + d) * HD + e];
  c1[t] = c;
}

// ---------------- phase 4: fused hid/beta/enhanced + output GEMM ----------------
__global__ __launch_bounds__(256) void k_fused(
    const float* __restrict__ E, const float* __restrict__ W1,
    const float* __restrict__ W2, const float* __restrict__ b2,
    const float* __restrict__ Wout, const float* __restrict__ bout,
    const float* __restrict__ g_new, const float* __restrict__ c1,
    float* __restrict__ out_enh, float* __restrict__ out_beta) {
  __shared__ float sA[MTILE * DIM];   // 128 KB row tile (h, later enhanced)
  __shared__ float sGnew[NH * HD];
  __shared__ float sC1[NH * HD];
  __shared__ float sW2[NH * HD];
  __shared__ float sBeta[MTILE * NH];

  int t = threadIdx.x;
  int row0 = blockIdx.x * MTILE;

  // ---- stage 64x512 fp32 tile via Tensor Data Mover (one wave issues) ----
  if (t < 32) {
    unsigned long long gaddr = (unsigned long long)(uintptr_t)(E + (size_t)row0 * DIM);
    unsigned ldsa = (unsigned)(uintptr_t)(&sA[0]);  // LDS offset = addr[31:0]
    v4u g0;
    g0[0] = 1u;                                   // count=1, load descriptor
    g0[1] = ldsa;                                 // lds_addr (bytes)
    g0[2] = (unsigned)(gaddr & 0xFFFFFFFFu);      // global_addr[31:0]
    g0[3] = (unsigned)((gaddr >> 32) & 0x01FFFFFFu) | (2u << 30);  // addr[56:32] | type=2
    v8i g1;
    g1[0] = 0x20000;                 // workgroup_mask=0, data_size=2 (4B)
    g1[1] = (int)(32768u << 16);     // tensor_dim0[15:0]=32768
    g1[2] = (int)(1u << 16);         // tensor_dim0[31:16]=0, tensor_dim1[15:0]=1
    g1[3] = (int)(32768u << 16);     // tensor_dim1[31:16]=0, tile_dim0=32768
    g1[4] = 1;                       // tile_dim1=1, tile_dim2=0
    g1[5] = 32768;                   // tensor_dim0_stride[31:0]
    g1[6] = 0;
    g1[7] = 0;
    v4i gz4 = {0, 0, 0, 0};
    v8i gz8 = {0, 0, 0, 0, 0, 0, 0, 0};
    __builtin_amdgcn_tensor_load_to_lds(g0, g1, gz4, gz4, gz8, 0);
    __builtin_amdgcn_s_wait_tensorcnt(0);
  }
  for (int i = t; i < NH * HD; i += 256) {
    sGnew[i] = g_new[i];
    sC1[i] = c1[i];
    sW2[i] = W2[i];
  }
  __syncthreads();

  int w = t >> 5;  // wave id (wave32): == head id == output column chunk
  int lane = t & 31;
  int lo = lane & 15, hi = lane >> 4;
  const v8f vzero = {0.f, 0.f, 0.f, 0.f, 0.f, 0.f, 0.f, 0.f};

  // ---- per-head hidden GEMM (64x64x64) + beta partial reduction ----
  {
    v8f acc[4][4];  // [msub][nsub]
#pragma unroll
    for (int m = 0; m < 4; ++m)
#pragma unroll
      for (int nn = 0; nn < 4; ++nn) acc[m][nn] = vzero;
    const float* w1h = W1 + w * 2 * HD * HD;  // [128][64], first 64 K-rows
    for (int kk = 0; kk < HD; kk += 4) {
      int k = kk + 2 * hi;
      v2f a[4];
#pragma unroll
      for (int m = 0; m < 4; ++m) {
        a[m][0] = sA[(m * 16 + lo) * DIM + w * HD + k];
        a[m][1] = sA[(m * 16 + lo) * DIM + w * HD + k + 1];
      }
#pragma unroll
      for (int nn = 0; nn < 4; ++nn) {
        int ncol = nn * 16 + lo;
        v2f b;
        b[0] = w1h[k * HD + ncol];
        b[1] = w1h[(k + 1) * HD + ncol];
#pragma unroll
        for (int m = 0; m < 4; ++m)
          acc[m][nn] = __builtin_amdgcn_wmma_f32_16x16x4_f32(
              false, a[m], false, b, (short)0, acc[m][nn], false, false);
      }
    }
    // hid = relu(acc + c1); beta partial = hid . W2 (reduce over 64 cols)
#pragma unroll
    for (int m = 0; m < 4; ++m) {
#pragma unroll
      for (int r = 0; r < 8; ++r) {
        float s = 0.f;
#pragma unroll
        for (int nn = 0; nn < 4; ++nn) {
          int ncol = nn * 16 + lo;
          float v = acc[m][nn][r] + sC1[w * HD + ncol];
          v = v > 0.f ? v : 0.f;
          s += v * sW2[w * HD + ncol];
        }
        s += __shfl_xor(s, 8, 16);
        s += __shfl_xor(s, 4, 16);
        s += __shfl_xor(s, 2, 16);
        s += __shfl_xor(s, 1, 16);
        if (lo == 0) sBeta[(m * 16 + r + 8 * hi) * NH + w] = s;
      }
    }
  }
  __syncthreads();

  // ---- beta = sigmoid(+b2); beta_avg; enhanced in place ----
  for (int i = t; i < MTILE * NH; i += 256) {
    int h = i & 7;
    float x = sBeta[i] + b2[h];
    sBeta[i] = 1.f / (1.f + __expf(-x));
  }
  __syncthreads();
  if (t < MTILE) {
    float s = 0.f;
#pragma unroll
    for (int h = 0; h < NH; ++h) s += sBeta[t * NH + h];
    out_beta[row0 + t] = s * (1.f / NH);
  }
  {  // each thread rewrites 128 contiguous elements (2 head segments of one row)
    int base = t * 128;
    int m = base >> 9;
    int colb = base & 511;
#pragma unroll
    for (int j = 0; j < 2; ++j) {
      int h = (colb >> 6) + j;
      float beta = sBeta[m * NH + h];
      float omb = 1.f - beta;
#pragma unroll 8
      for (int d = 0; d < 64; ++d) {
        int c = colb + j * 64 + d;
        sA[m * DIM + c] = omb * sA[m * DIM + c] + beta * sGnew[c];
      }
    }
  }
  __syncthreads();

  // ---- output GEMM: out[64x512] = enhanced(64x512) @ Wout(512x512) + bout ----
  {
    v8f acc[4][4];  // [msub][nsub]
#pragma unroll
    for (int m = 0; m < 4; ++m)
#pragma unroll
      for (int nn = 0; nn < 4; ++nn) acc[m][nn] = vzero;
    int col0 = w * 64;
    for (int kk = 0; kk < DIM; kk += 4) {
      int k = kk + 2 * hi;
      v2f a[4];
#pragma unroll
      for (int m = 0; m < 4; ++m) {
        a[m][0] = sA[(m * 16 + lo) * DIM + k];
        a[m][1] = sA[(m * 16 + lo) * DIM + k + 1];
      }
      if (kk + 16 < DIM)
        __builtin_prefetch(&Wout[(size_t)(kk + 16) * DIM + col0 + lo], 0, 1);
#pragma unroll
      for (int nn = 0; nn < 4; ++nn) {
        int ncol = col0 + nn * 16 + lo;
        v2f b;
        b[0] = Wout[(size_t)k * DIM + ncol];
        b[1] = Wout[(size_t)(k + 1) * DIM + ncol];
#pragma unroll
        for (int m = 0; m < 4; ++m)
          acc[m][nn] = __builtin_amdgcn_wmma_f32_16x16x4_f32(
              false, a[m], false, b, (short)0, acc[m][nn], false, false);
      }
    }
#pragma unroll
    for (int nn = 0; nn < 4; ++nn) {
      int ncol = col0 + nn * 16 + lo;
      float bv = bout[ncol];
#pragma unroll
      for (int m = 0; m < 4; ++m) {
#pragma unroll
        for (int r = 0; r < 8; ++r) {
          int grow = row0 + m * 16 + r + 8 * hi;
          out_enh[(size_t)grow * DIM + ncol] = acc[m][nn][r] + bv;
        }
      }
    }
  }
}

extern "C" void kernel_launch(void* const* d_in, const int* in_sizes, int n_in,
                              void* d_out, int out_size, void* d_ws, size_t ws_size,
                              hipStream_t stream) {
  const float* E    = (const float*)d_in[0];  // entity_embeds (N,512)
  const float* gmem = (const float*)d_in[1];  // global_memories (8,64)
  const float* attn = (const float*)d_in[2];  // attn_vec (8,128)
  const float* W1   = (const float*)d_in[3];  // (8,128,64)
  const float* b1   = (const float*)d_in[4];  // (8,64)
  const float* W2   = (const float*)d_in[5];  // (8,64)
  const float* b2   = (const float*)d_in[6];  // (8,)
  const float* Wout = (const float*)d_in[7];  // (512,512)
  const float* bout = (const float*)d_in[8];  // (512,)

  float* out_enh  = (float*)d_out;
  float* out_beta = out_enh + (size_t)NROW * DIM;

  float* ws       = (float*)d_ws;
  float* ws_e     = ws;                                  // N*8 floats
  unsigned* gmax  = (unsigned*)(ws + (size_t)NROW * NH); // 8
  float* gsum     = (float*)(gmax + NH);                 // 8
  float* G        = gsum + NH;                           // 512
  float* g_new    = G + NH * HD;                         // 512
  float* c1       = g_new + NH * HD;                     // 512

  k_init<<<1, 512, 0, stream>>>(gmax, gsum, G);
  k_e<<<(NROW * NH) / 256, 256, 0, stream>>>(E, gmem, attn, ws_e, gmax);
  k_sum<<<(NROW * NH) / 256, 256, 0, stream>>>(E, ws_e, gmax, gsum, G);
  k_final<<<1, 512, 0, stream>>>(gsum, G, W1, b1, g_new, c1);
  k_fused<<<NROW / MTILE, 256, 0, stream>>>(E, W1, W2, b2, Wout, bout, g_new,
                                            c1, out_enh, out_beta);
}